// Q99FP4Quantizer_27951647163126
// MI455X (gfx1250) — compile-verified
//
#include <hip/hip_runtime.h>

// FP4-codebook STE quantizer, forward value == hard_quantize(x).
// Memory-bound streaming kernel: 64MiB in + 64MiB out @ 23.3TB/s ~= 5.8us floor.
// Per-element cost kept to ~10 VALU + 1 DS so VALU issue stays under the HBM roof.

__device__ __forceinline__ float nearest_cb(const float* cbs, float v) {
    // Reference argmin semantics: strict '<' keeps the FIRST (lowest-index) value on ties.
    float best = cbs[0];
    float bd = fabsf(v - cbs[0]);
#pragma unroll
    for (int j = 1; j < 16; ++j) {
        float d = fabsf(v - cbs[j]);
        if (d < bd) { bd = d; best = cbs[j]; }
    }
    return best;
}

__device__ __forceinline__ float quant_one(float x, float s8, const float2* lut) {
    // a8 == 8*|x*scale| exactly (scaling by 8 is exact in fp32).
    float a8 = fabsf(x) * s8;              // v_mul_f32 with |src| modifier
    unsigned c = (unsigned)a8;             // v_cvt_u32_f32 (truncate)
    c = c < 48u ? c : 48u;                 // v_min_u32
    unsigned neg = __float_as_uint(x) >> 31;
    float2 pr = lut[(neg << 6) + c];       // ds_load_b64
    bool exact = (a8 == (float)c);         // v_cvt_f32_u32 + v_cmp_eq_f32
    return exact ? pr.y : pr.x;            // v_cndmask (tie-break on exact boundary)
}

__global__ void __launch_bounds__(256)
fp4_ste_quant_kernel(const float4* __restrict__ xv, const float* __restrict__ cb,
                     const float* __restrict__ sp, float4* __restrict__ ov, int nv) {
    __shared__ float  cbs[16];
    __shared__ float2 lut[128];            // [0..48]=positive table, [64..112]=negative
    const int tid = threadIdx.x;

    // --- gfx1250 async path: copy the 64B codebook to LDS via GLOBAL_LOAD_ASYNC_TO_LDS ---
    if (tid < 8) {
        unsigned lds_off = (unsigned)(unsigned long long)(const void*)&cbs[tid * 2];
        unsigned long long ga = (unsigned long long)(const void*)(cb + tid * 2);
        asm volatile("global_load_async_to_lds_b64 %0, %1, off"
                     :: "v"(lds_off), "v"(ga) : "memory");
    }
    asm volatile("s_wait_asynccnt 0" ::: "memory");
    __syncthreads();

    const float scale = sp[0];

    // --- Build the 2x49-cell LUT by evaluating the reference quantizer itself ---
    // Each cell stores: .x = value for cell interior, .y = value at exact left endpoint
    // (argmin tie rule handled automatically by nearest_cb's first-match-wins scan).
    if (tid < 98) {
        int k   = tid >> 1;                // cell 0..48  (cell width = 1/8 in |xs|)
        int neg = tid & 1;
        float am  = ((float)k + 0.5f) * 0.125f;   // interior representative
        float as  = (float)k * 0.125f;            // exact left endpoint
        float xm  = neg ? -am : am;
        float xst = neg ? -as : as;
        float qx = nearest_cb(cbs, xm)  / scale;  // real division: bit-faithful q/scale
        float qy = nearest_cb(cbs, xst) / scale;
        lut[(neg << 6) + k] = make_float2(qx, qy);
    }
    __syncthreads();

    // --- Streaming loop: b128 loads/stores, 4 elements per thread per iteration ---
    const float s8 = scale * 8.0f;
    const int stride = gridDim.x * blockDim.x;
    for (int i = blockIdx.x * blockDim.x + tid; i < nv; i += stride) {
        float4 v = xv[i];
        float4 r;
        r.x = quant_one(v.x, s8, lut);
        r.y = quant_one(v.y, s8, lut);
        r.z = quant_one(v.z, s8, lut);
        r.w = quant_one(v.w, s8, lut);
        ov[i] = r;
    }
}

extern "C" void kernel_launch(void* const* d_in, const int* in_sizes, int n_in,
                              void* d_out, int out_size, void* d_ws, size_t ws_size,
                              hipStream_t stream) {
    const float* x  = (const float*)d_in[0];   // [4096*4096] fp32
    const float* cb = (const float*)d_in[1];   // [16] sorted fp32 codebook
    const float* sp = (const float*)d_in[2];   // [1] scale
    float* out = (float*)d_out;

    int n  = in_sizes[0];
    int nv = n >> 2;                           // 4096^2 is divisible by 4
    int block = 256;                           // 8 wave32s per workgroup
    long long want = ((long long)nv + block - 1) / block;
    int grid = (int)(want < 2048 ? (want < 1 ? 1 : want) : 2048);

    fp4_ste_quant_kernel<<<grid, block, 0, stream>>>(
        (const float4*)x, cb, sp, (float4*)out, nv);
}